// WHT2Layer_15522011808172
// MI455X (gfx1250) — compile-verified
//
#include <hip/hip_runtime.h>

typedef __attribute__((ext_vector_type(2))) float v2f;
typedef __attribute__((ext_vector_type(8))) float v8f;

// One wave handles 16 columns (16 consecutive a3) of one (i0, q) slab, and the
// full 64-point WHT along the transform axis (stride St, compile-time).
//   element address = i0*S0 + q*Sq + t*St + a3,  S0 = 64*64*128
// Decomposition: H64 = H16 (x) H4, t = hi*4 + lo.
//   Step 1 (VALU): radix-4 butterfly over lo  -> Z[hi, lo']
//   Step 2 (WMMA): Y[hi',lo'] = sum_hi H16[hi',hi] * Z[hi,lo']
//                  as 4 chained v_wmma_f32_16x16x4_f32 per lo'.
// Sq/St are template parameters so every t*St folds into the 24-bit IOFFSET
// of global_load_b32/global_store_b32 (max 63*8192*4 B < 8 MiB): one base
// address VGPR, zero per-access address arithmetic.
template <int Sq, int St>
__global__ __launch_bounds__(256) void fwht64_pass(
    const float* __restrict__ src, float* __restrict__ dst)
{
    const int lane = threadIdx.x & 31;
    const int w    = blockIdx.x * 8 + (threadIdx.x >> 5);   // 0..32767
    const int a3b  = (w & 7) << 4;        // a3 base (8 groups of 16)
    const int q    = (w >> 3) & 63;       // non-transform middle axis
    const int i0   = w >> 9;              // 0..63

    const int n    = lane & 15;           // matmul column (a3 within group)
    const int half = lane >> 4;           // 0: lanes 0-15, 1: lanes 16-31

    const int S0 = 64 * 64 * 128;
    const int colbase = i0 * S0 + q * Sq + a3b + n;
    const float* __restrict__ srcc = src + colbase;
    float* __restrict__       dstc = dst + colbase;

    // ---- Step 1: load 32 values per lane, radix-4 butterfly on lo bits ----
    // Lane needs hi-groups: half==0 -> {0,1,4,5,8,9,12,13}; half==1 -> +2.
    float z[8][4];
#pragma unroll
    for (int g = 0; g < 8; ++g) {
        const int hi = ((g >> 1) << 2) + (g & 1) + (half << 1);
        float x0 = srcc[(hi * 4 + 0) * St];
        float x1 = srcc[(hi * 4 + 1) * St];
        float x2 = srcc[(hi * 4 + 2) * St];
        float x3 = srcc[(hi * 4 + 3) * St];
        float s0 = x0 + x1, s1 = x0 - x1;
        float s2 = x2 + x3, s3 = x2 - x3;
        z[g][0] = s0 + s2;   // lo' = 0: ++++
        z[g][1] = s1 + s3;   // lo' = 1: +-+-
        z[g][2] = s0 - s2;   // lo' = 2: ++--
        z[g][3] = s1 - s3;   // lo' = 3: +--+
    }

    // ---- A-matrix: H16 column blocks, per ISA 16x4 f32 A layout ----
    // a.x: K = 4kc + 2*half ; a.y: K = 4kc + 1 + 2*half ; M = lane&15.
    v2f A[4];
#pragma unroll
    for (int kc = 0; kc < 4; ++kc) {
        const int k0 = 4 * kc + 2 * half;
        const int k1 = k0 + 1;
        A[kc].x = (__builtin_popcount(n & k0) & 1) ? -1.0f : 1.0f;
        A[kc].y = (__builtin_popcount(n & k1) & 1) ? -1.0f : 1.0f;
    }

    // ---- Step 2: per lo', 16x16x16 matmul as 4 chained 16x16x4 WMMAs ----
#pragma unroll
    for (int lp = 0; lp < 4; ++lp) {
        v8f acc = {};
#pragma unroll
        for (int kc = 0; kc < 4; ++kc) {
            // B (4x16) layout: b.x row = 4kc+2*half, b.y row = 4kc+1+2*half,
            // column = lane&15.  Those rows live in z[2kc], z[2kc+1].
            v2f B;
            B.x = z[2 * kc][lp];
            B.y = z[2 * kc + 1][lp];
            acc = __builtin_amdgcn_wmma_f32_16x16x4_f32(
                /*neg_a=*/false, A[kc], /*neg_b=*/false, B,
                /*c_mod=*/(short)0, acc, /*reuse_a=*/false, /*reuse_b=*/false);
        }
        // D layout: acc[r] = Y[M = r + 8*half, N = n]  ->  t = M*4 + lp
#pragma unroll
        for (int r = 0; r < 8; ++r) {
            const int t = (r + 8 * half) * 4 + lp;
            dstc[t * St] = acc[r];
        }
    }
}

extern "C" void kernel_launch(void* const* d_in, const int* in_sizes, int n_in,
                              void* d_out, int out_size, void* d_ws, size_t ws_size,
                              hipStream_t stream) {
    (void)in_sizes; (void)n_in; (void)out_size; (void)d_ws; (void)ws_size;

    const float* x  = (const float*)d_in[0];
    float*       t1 = (float*)d_out;                                  // FWHT axis 1
    float*       t2 = (float*)d_out + (size_t)64 * 64 * 64 * 128;     // then axis 2

    const dim3 grid(4096);   // 4096 blocks * 8 waves = 32768 waves = 524288 cols / 16
    const dim3 block(256);

    // Pass 1: transform along a1 (St = 64*128), q = a2 (Sq = 128)
    fwht64_pass<128, 8192><<<grid, block, 0, stream>>>(x, t1);
    // Pass 2: transform along a2 of t1 (St = 128), q = a1 (Sq = 8192)
    fwht64_pass<8192, 128><<<grid, block, 0, stream>>>(t1, t2);
}